// Causal_Temporal_Map_Attention_50362786513377
// MI455X (gfx1250) — compile-verified
//
#include <hip/hip_runtime.h>

typedef float v2f __attribute__((ext_vector_type(2)));
typedef float v8f __attribute__((ext_vector_type(8)));

#define BATCH    4
#define NROWS    4096
#define TDIM     256
#define GCHUNK   512
#define GNCHUNKS (NROWS / GCHUNK)   // 8

#define WMMA_F32(A, B, C) \
    __builtin_amdgcn_wmma_f32_16x16x4_f32(false, (A), false, (B), (short)0, (C), false, false)

// d_ws layout: G[4][256][256] f32 (1 MB) then M[4][256][256] f32 (1 MB).

__global__ __launch_bounds__(256) void zero_ws_kernel(float* __restrict__ g) {
    g[blockIdx.x * 256 + threadIdx.x] = 0.0f;
}

// G_b = X_b^T @ X_b. One wave computes a 32x32 region (2x2 WMMA tiles) per
// 512-row chunk; G is symmetric so only regions si<=ti are computed, with
// off-diagonal regions mirrored. fp32 atomics accumulate chunk partials.
__global__ __launch_bounds__(32) void gram_kernel(const float* __restrict__ X,
                                                  float* __restrict__ G) {
    const int si = blockIdx.x >> 3;         // 0..7 region row
    const int ti = blockIdx.x & 7;          // 0..7 region col
    if (si > ti) return;                    // uniform exit: EXEC stays all-ones below
    const int chunk = blockIdx.y;           // 0..7
    const int b     = blockIdx.z;           // 0..3
    const int sBase = si << 5;
    const int tBase = ti << 5;
    const int lane  = threadIdx.x & 15;
    const int half  = threadIdx.x >> 4;     // K pair {0,1} vs {2,3}

    const float* Xb = X + (size_t)b * NROWS * TDIM;
    const int n0 = chunk * GCHUNK + 2 * half;

    v8f acc00 = {}, acc01 = {}, acc10 = {}, acc11 = {};
    for (int kk = 0; kk < GCHUNK; kk += 4) {
        const float* r0 = Xb + (size_t)(n0 + kk) * TDIM;
        const float* r1 = r0 + TDIM;
        v2f a0, a1, b0, b1;
        a0.x = r0[sBase + lane];        a0.y = r1[sBase + lane];
        a1.x = r0[sBase + 16 + lane];   a1.y = r1[sBase + 16 + lane];
        b0.x = r0[tBase + lane];        b0.y = r1[tBase + lane];
        b1.x = r0[tBase + 16 + lane];   b1.y = r1[tBase + 16 + lane];
        acc00 = WMMA_F32(a0, b0, acc00);
        acc01 = WMMA_F32(a0, b1, acc01);
        acc10 = WMMA_F32(a1, b0, acc10);
        acc11 = WMMA_F32(a1, b1, acc11);
    }

    float* Gb = G + (size_t)b * TDIM * TDIM;
    const bool mirror = (si != ti);   // diagonal region already covers its transpose
#pragma unroll
    for (int q = 0; q < 4; ++q) {
        const int rb = sBase + ((q >> 1) << 4);
        const int cb = tBase + ((q & 1) << 4);
        const v8f acc = (q == 0) ? acc00 : (q == 1) ? acc01 : (q == 2) ? acc10 : acc11;
#pragma unroll
        for (int r = 0; r < 8; ++r) {
            const int row = rb + r + 8 * half;
            const int col = cb + lane;
            __hip_atomic_fetch_add(&Gb[row * TDIM + col], acc[r],
                                   __ATOMIC_RELAXED, __HIP_MEMORY_SCOPE_AGENT);
            if (mirror)
                __hip_atomic_fetch_add(&Gb[col * TDIM + row], acc[r],
                                       __ATOMIC_RELAXED, __HIP_MEMORY_SCOPE_AGENT);
        }
    }
}

// M_b[i][j] = sum_{k < i} W[k][i] * G_b[k][j]  (Wm = triu(W,1), transposed).
// Tiny; VALU with uniform triangular loop bound, j coalesced, W broadcast.
__global__ __launch_bounds__(256) void wmask_gemm_kernel(const float* __restrict__ W,
                                                         const float* __restrict__ G,
                                                         float* __restrict__ M) {
    const int j = threadIdx.x;
    const int i = blockIdx.x;
    const int b = blockIdx.y;
    const float* Gb = G + (size_t)b * TDIM * TDIM;
    float acc = 0.0f;
    for (int k = 0; k < i; ++k) {
        acc = fmaf(W[k * TDIM + i], Gb[k * TDIM + j], acc);
    }
    M[((size_t)b * TDIM + i) * TDIM + j] = acc;
}

// out_b = X_b @ M_b. One wave computes a 32x32 output block (2x2 WMMA tiles),
// K = 256 in steps of 4; A fragments are float2 loads, each fragment reused 2x.
__global__ __launch_bounds__(32) void out_gemm_kernel(const float* __restrict__ X,
                                                      const float* __restrict__ M,
                                                      float* __restrict__ Out) {
    const int nBase = blockIdx.x << 5;      // 0..4064 step 32
    const int tBase = blockIdx.y << 5;      // 0..224  step 32
    const int b     = blockIdx.z;
    const int lane  = threadIdx.x & 15;
    const int half  = threadIdx.x >> 4;

    const float* Xb = X + (size_t)b * NROWS * TDIM;
    const float* Mb = M + (size_t)b * TDIM * TDIM;
    const float* xrow0 = Xb + (size_t)(nBase + lane) * TDIM + 2 * half;
    const float* xrow1 = xrow0 + (size_t)16 * TDIM;

    v8f acc00 = {}, acc01 = {}, acc10 = {}, acc11 = {};
    for (int k0 = 0; k0 < TDIM; k0 += 4) {
        v2f a0 = *(const v2f*)(xrow0 + k0);              // X[nBase+lane][k0+2h..+1]
        v2f a1 = *(const v2f*)(xrow1 + k0);              // X[nBase+16+lane][...]
        const float* mrow = Mb + (size_t)(k0 + 2 * half) * TDIM;
        v2f b0, b1;
        b0.x = mrow[tBase + lane];        b0.y = mrow[TDIM + tBase + lane];
        b1.x = mrow[tBase + 16 + lane];   b1.y = mrow[TDIM + tBase + 16 + lane];
        acc00 = WMMA_F32(a0, b0, acc00);
        acc01 = WMMA_F32(a0, b1, acc01);
        acc10 = WMMA_F32(a1, b0, acc10);
        acc11 = WMMA_F32(a1, b1, acc11);
    }

    float* Ob = Out + (size_t)b * NROWS * TDIM;
#pragma unroll
    for (int r = 0; r < 8; ++r) {
        const int row0 = nBase + r + 8 * half;
        const int row1 = row0 + 16;
        Ob[(size_t)row0 * TDIM + tBase + lane]      = acc00[r];
        Ob[(size_t)row0 * TDIM + tBase + 16 + lane] = acc01[r];
        Ob[(size_t)row1 * TDIM + tBase + lane]      = acc10[r];
        Ob[(size_t)row1 * TDIM + tBase + 16 + lane] = acc11[r];
    }
}

extern "C" void kernel_launch(void* const* d_in, const int* in_sizes, int n_in,
                              void* d_out, int out_size, void* d_ws, size_t ws_size,
                              hipStream_t stream) {
    const float* x = (const float*)d_in[0];   // [4, 4096, 256]
    const float* W = (const float*)d_in[1];   // [256, 256]
    float* out = (float*)d_out;               // [4, 4096, 256]

    float* G = (float*)d_ws;                        // 4*256*256 f32
    float* M = G + (size_t)BATCH * TDIM * TDIM;     // 4*256*256 f32

    zero_ws_kernel<<<(BATCH * TDIM * TDIM) / 256, 256, 0, stream>>>(G);
    gram_kernel<<<dim3(64, GNCHUNKS, BATCH), 32, 0, stream>>>(x, G);
    wmask_gemm_kernel<<<dim3(TDIM, BATCH), TDIM, 0, stream>>>(W, G, M);
    out_gemm_kernel<<<dim3(NROWS / 32, TDIM / 32, BATCH), 32, 0, stream>>>(x, M, out);
}